// First_level_net_41678362640734
// MI455X (gfx1250) — compile-verified
//
#include <hip/hip_runtime.h>

typedef __attribute__((ext_vector_type(2))) float v2f;
typedef __attribute__((ext_vector_type(8))) float v8f;
typedef __attribute__((ext_vector_type(4))) unsigned int v4u;
typedef __attribute__((ext_vector_type(4))) int v4i;
typedef __attribute__((ext_vector_type(8))) int v8i;

#define TH   0.01f
#define HW   2046          // conv output height/width
#define XW   2048          // input height/width
#define TILE 30            // output tile (even, for 2x2 pooling)
#define FT   31            // f tile = TILE + 1 halo
#define XT   33            // x tile = FT + 2
#define PD   1023          // pooled dim

__global__ __launch_bounds__(256) void flnet_fused_kernel(
    const float* __restrict__ x, const float* __restrict__ w,
    float* __restrict__ out) {
  __shared__ float sx[XT][XT];          // x tile (TDM packs rows contiguously)
  __shared__ float sf[6][FT][32];       // relu(conv) tile, 6 channels
  __shared__ float sm[12][TILE][TILE];  // mult_x tile for pooling

  const int tid = threadIdx.x;
  const int gx0 = blockIdx.x * TILE;
  const int gy0 = blockIdx.y * TILE;

  // ---- Stage 1: x tile -> LDS via Tensor Data Mover (one DMA per block).
  // TDM zero-fills out-of-tensor reads; zero-filled halo feeds only f entries
  // that are never consumed. Fallback: manual clamped loads. ----
#if defined(__AMDGCN__) && __has_builtin(__builtin_amdgcn_tensor_load_to_lds)
  if (tid < 32) {                                  // single wave issues the DMA
    unsigned int lds_base = (unsigned int)(unsigned long long)&sx[0][0];
    unsigned long long gaddr = (unsigned long long)x +
        4ull * ((unsigned long long)gy0 * XW + (unsigned long long)gx0);
    // D# group0: count=1 | lds_addr | global_addr[56:0] | type=2
    v4u g0 = { 1u,
               lds_base,
               (unsigned int)(gaddr & 0xffffffffu),
               (unsigned int)((gaddr >> 32) & 0x01ffffffu) | (2u << 30) };
    // D# group1: data_size=4B(code 2); tensor 2048x2048; tile 33x33;
    // row stride 2048; dim1 stride 2048*2048 (unused for 2D tile).
    v8i g1 = { (int)(2u << 16),                       // data_size
               (int)((XW & 0xffffu) << 16),           // tensor_dim0[15:0]<<16
               (int)((XW & 0xffffu) << 16),           // dim0[31:16]=0 | dim1[15:0]<<16
               (int)(((unsigned)XT) << 16),           // dim1[31:16]=0 | tile_dim0<<16
               (int)XT,                               // tile_dim1 | tile_dim2=0
               (int)XW,                               // tensor_dim0_stride[31:0]
               0,                                     // stride0[47:32] | stride1[15:0]
               (int)((4ull * XW * XW) >> 32 << 0) };  // placeholder, see below
    g1[7] = (int)(((unsigned long long)XW * XW) >> 16);  // stride1[47:16]
    v4i gz4 = { 0, 0, 0, 0 };
#if __clang_major__ >= 23
    v8i gz8 = { 0, 0, 0, 0, 0, 0, 0, 0 };
    __builtin_amdgcn_tensor_load_to_lds(g0, g1, gz4, gz4, gz8, 0);
#else
    __builtin_amdgcn_tensor_load_to_lds(g0, g1, gz4, gz4, 0);
#endif
    __builtin_amdgcn_s_wait_tensorcnt(0);
  }
#else
  for (int idx = tid; idx < XT * XT; idx += 256) {
    int r = idx / XT, c = idx % XT;
    int gr = gy0 + r; if (gr > XW - 1) gr = XW - 1;
    int gc = gx0 + c; if (gc > XW - 1) gc = XW - 1;
    sx[r][c] = x[gr * XW + gc];
  }
#endif

  // ---- Stage 2: weights into WMMA A-operand layout (16x4 f32, ISA 7.12.2):
  // lanes 0-15: M=lane, VGPR0/1 = K0/K1 ; lanes 16-31: M=lane-16, K2/K3 ----
  const int lane = tid & 31;
  const int wid  = tid >> 5;
  const int chl  = lane & 15;
  const int koff = (lane < 16) ? 0 : 2;
  v2f A[3];
#pragma unroll
  for (int t = 0; t < 3; ++t) {
    int k0 = 4 * t + koff, k1 = k0 + 1;
    A[t][0] = (chl < 6 && k0 < 9) ? w[chl * 9 + k0] : 0.0f;
    A[t][1] = (chl < 6 && k1 < 9) ? w[chl * 9 + k1] : 0.0f;
  }

  __syncthreads();

  // ---- Stage 3: conv+relu via V_WMMA_F32_16X16X4_F32 (K=9 padded to 12,
  // 3 WMMAs per 16 pixels; D = 16 channels x 16 pixels) ----
  for (int g = wid; g < FT * 2; g += 8) {   // wave-uniform loop, EXEC all-1s
    int row = g >> 1;
    int p = (g & 1) * 16 + chl;
    if (p > FT - 1) p = FT - 1;             // clamp (duplicate col, same value)
    v8f acc = {};
#pragma unroll
    for (int t = 0; t < 3; ++t) {
      int k0 = 4 * t + koff, k1 = k0 + 1;
      v2f b;
      b[0] = (k0 < 9) ? sx[row + k0 / 3][p + k0 % 3] : 0.0f;
      b[1] = (k1 < 9) ? sx[row + k1 / 3][p + k1 % 3] : 0.0f;
      acc = __builtin_amdgcn_wmma_f32_16x16x4_f32(
          false, A[t], false, b, (short)0, acc, false, false);
    }
    if (lane < 16) {                        // lanes 0-15 hold channels 0..7
#pragma unroll
      for (int v = 0; v < 6; ++v) {
        float z = acc[v];
        sf[v][row][p] = z > 0.0f ? z : 0.0f;   // relu (== abs(relu))
      }
    }
  }

  __syncthreads();

  // ---- Stage 4: shifted elementwise products -> sm (LDS) + global mult_x.
  // shift0/shift1 semantics: result[i]=a[i+1] (i<H-2), 0 at H-2, a[H-1] at H-1.
  const long plane = (long)HW * HW;
  for (int idx = tid; idx < TILE * TILE; idx += 256) {
    int r = idx / TILE, c = idx % TILE;
    int gi = gy0 + r, gj = gx0 + c;
    if (gi >= HW || gj >= HW) continue;
    int  r2 = (gi == HW - 1) ? r : r + 1;
    int  c2 = (gj == HW - 1) ? c : c + 1;
    bool zi = (gi == HW - 2);
    bool zj = (gj == HW - 2);

    float P[6], S0[6], S1[6], S01[6];
#pragma unroll
    for (int ch = 0; ch < 6; ++ch) {
      P[ch]      = sf[ch][r][c];
      float fdn  = sf[ch][r2][c];
      float frt  = sf[ch][r][c2];
      float fdg  = sf[ch][r2][c2];
      S0[ch]  = zi ? 0.0f : fdn;               // shift0
      S1[ch]  = zj ? 0.0f : frt;               // shift1
      S01[ch] = (zi || zj) ? 0.0f : fdg;       // shift1(shift0) == shift0(shift1)
    }

    float m_fd  = (P[4] * S1[4] <= TH) ? 1.0f : 0.0f;
    float m_sd  = (P[5] * S1[5] <= TH) ? 1.0f : 0.0f;
    float m_fdd = (P[4] * S0[4] <= TH) ? 1.0f : 0.0f;
    float m_sdd = (P[5] * S0[5] <= TH) ? 1.0f : 0.0f;

    float* outp = out + (long)gi * HW + gj;
#pragma unroll
    for (int brn = 0; brn < 2; ++brn) {
      int vi = brn, hi = brn + 2;              // (V1,H1) then (oV1,oH1)
      float vert  = S0[vi]  * P[vi];
      float fd    = S01[vi] * P[vi];
      float sd    = S0[vi]  * S1[vi];
      float horiz = S1[hi]  * P[hi];
      float fdd   = S01[hi] * P[hi];
      float sdd   = S0[hi]  * S1[hi];
      float o0 = horiz * ((vert  <= TH) ? 1.0f : 0.0f);
      float o1 = fd * m_fd;
      float o2 = vert  * ((horiz <= TH) ? 1.0f : 0.0f);
      float o3 = sd * m_sd;
      float o4 = fdd * m_fdd;
      float o5 = sdd * m_sdd;
      int cb = brn * 6;
      sm[cb + 0][r][c] = o0; __builtin_nontemporal_store(o0, outp + (cb + 0) * plane);
      sm[cb + 1][r][c] = o1; __builtin_nontemporal_store(o1, outp + (cb + 1) * plane);
      sm[cb + 2][r][c] = o2; __builtin_nontemporal_store(o2, outp + (cb + 2) * plane);
      sm[cb + 3][r][c] = o3; __builtin_nontemporal_store(o3, outp + (cb + 3) * plane);
      sm[cb + 4][r][c] = o4; __builtin_nontemporal_store(o4, outp + (cb + 4) * plane);
      sm[cb + 5][r][c] = o5; __builtin_nontemporal_store(o5, outp + (cb + 5) * plane);
    }
  }

  __syncthreads();

  // ---- Stage 5: 2x2 average pool from LDS ----
  const long mxsize = 12L * HW * HW;
  for (int idx = tid; idx < 12 * 15 * 15; idx += 256) {
    int ch = idx / 225;
    int rem = idx % 225;
    int pr = rem / 15, pc = rem % 15;
    int pi = gy0 / 2 + pr, pj = gx0 / 2 + pc;
    if (pi >= PD || pj >= PD) continue;
    float s = sm[ch][2 * pr][2 * pc]     + sm[ch][2 * pr][2 * pc + 1] +
              sm[ch][2 * pr + 1][2 * pc] + sm[ch][2 * pr + 1][2 * pc + 1];
    __builtin_nontemporal_store(
        0.25f * s, out + mxsize + (long)ch * PD * PD + (long)pi * PD + pj);
  }
}

extern "C" void kernel_launch(void* const* d_in, const int* in_sizes, int n_in,
                              void* d_out, int out_size, void* d_ws, size_t ws_size,
                              hipStream_t stream) {
  const float* x = (const float*)d_in[0];   // [1,1,2048,2048]
  const float* w = (const float*)d_in[1];   // [6,1,3,3]
  float* out = (float*)d_out;               // mult_x (12*2046^2) ++ pooled (12*1023^2)
  (void)in_sizes; (void)n_in; (void)out_size; (void)d_ws; (void)ws_size;
  dim3 grid((HW + TILE - 1) / TILE, (HW + TILE - 1) / TILE);
  flnet_fused_kernel<<<grid, 256, 0, stream>>>(x, w, out);
}